// Cross_modal_Center_ContrastiveLoss2_36618891166026
// MI455X (gfx1250) — compile-verified
//
#include <hip/hip_runtime.h>
#include <math.h>

#define NUM_CLASSES 395
#define CPAD        400              // padded to multiple of 16 (25 tiles)
#define FEAT        512
#define NSAMP       8192
#define NT          25               // CPAD/16
#define NTILES      (NT*NT)          // 625
#define MARGIN_F    0.5f

typedef __attribute__((ext_vector_type(2))) float v2f;
typedef __attribute__((ext_vector_type(8))) float v8f;

// workspace layout (floats)
#define OFF_PPAD 0
#define OFF_SQ   (CPAD*FEAT)             // 204800
#define OFF_CNT  (OFF_SQ  + CPAD)        // 205200
#define OFF_PL1  (OFF_CNT + CPAD)        // 205600
#define OFF_PL2  (OFF_PL1 + CPAD)        // 206000
#define OFF_PL3  (OFF_PL2 + CPAD)        // 206400  (+625) -> ~828 KB total

__device__ __forceinline__ float smooth_l1_elem(float x) {
    float d = fabsf(x);
    return (d < 1.f) ? 0.5f * d * d : d - 0.5f;
}

// ---- zero-padded copy of centers_param into ws ----
__global__ void pad_kernel(const float* __restrict__ cp, float* __restrict__ ppad) {
    int idx = blockIdx.x * blockDim.x + threadIdx.x;   // < CPAD*FEAT
    int row = idx / FEAT;
    ppad[idx] = (row < NUM_CLASSES) ? cp[idx] : 0.f;
}

// ---- per-row squared norms of padded centers ----
__global__ void sq_kernel(const float* __restrict__ ppad, float* __restrict__ sq) {
    __shared__ float sm[256];
    int r = blockIdx.x, t = threadIdx.x;
    const float* p = ppad + (size_t)r * FEAT;
    float s = p[t] * p[t] + p[t + 256] * p[t + 256];
    sm[t] = s; __syncthreads();
    for (int o = 128; o > 0; o >>= 1) { if (t < o) sm[t] += sm[t + o]; __syncthreads(); }
    if (t == 0) sq[r] = sm[0];
}

// ---- per-class: counts, and loss1/loss2 partials (counts-weighted SmoothL1 sums) ----
__global__ void class_kernel(const float* __restrict__ m1, const float* __restrict__ m2,
                             const float* __restrict__ cp, const int* __restrict__ tg,
                             float* __restrict__ cnts, float* __restrict__ pl1,
                             float* __restrict__ pl2) {
    __shared__ float sm1[256], sm2[256];
    int c = blockIdx.x, t = threadIdx.x;
    float aR0 = 0.f, aR1 = 0.f, aT0 = 0.f, aT1 = 0.f, cnt = 0.f;
    if (c < NUM_CLASSES) {
        for (int i = 0; i < NSAMP; ++i) {
            if (tg[i] == c) {                       // uniform per block -> scalar branch
                cnt += 1.f;
                const float* r = m1 + (size_t)i * FEAT;
                const float* q = m2 + (size_t)i * FEAT;
                aR0 += r[t]; aR1 += r[t + 256];
                aT0 += q[t]; aT1 += q[t + 256];
            }
        }
    }
    float inv = 1.f / fmaxf(cnt, 1.f);
    float c0 = 0.f, c1 = 0.f;
    if (c < NUM_CLASSES) {
        c0 = cp[(size_t)c * FEAT + t];
        c1 = cp[(size_t)c * FEAT + t + 256];
    }
    float p1 = cnt * (smooth_l1_elem(aR0 * inv - c0) + smooth_l1_elem(aR1 * inv - c1));
    float p2 = cnt * (smooth_l1_elem(aT0 * inv - c0) + smooth_l1_elem(aT1 * inv - c1));
    sm1[t] = p1; sm2[t] = p2; __syncthreads();
    for (int o = 128; o > 0; o >>= 1) {
        if (t < o) { sm1[t] += sm1[t + o]; sm2[t] += sm2[t + o]; }
        __syncthreads();
    }
    if (t == 0) { cnts[c] = cnt; pl1[c] = sm1[0]; pl2[c] = sm2[0]; }
}

// ---- Gram tiles via fp32 WMMA + fused hinge^2 weighting; one wave per 16x16 tile ----
__global__ __launch_bounds__(32) void gram_kernel(const float* __restrict__ ppad,
                                                  const float* __restrict__ sq,
                                                  const float* __restrict__ cnts,
                                                  float* __restrict__ pl3) {
    int tile = blockIdx.x;
    int ti = tile / NT, tj = tile % NT;
    int lane = threadIdx.x;
    int half = lane >> 4;            // 0: K=0,1 ; 1: K=2,3   (ISA 16x4 f32 A layout)
    int l    = lane & 15;

    const float* Arow = ppad + (size_t)(ti * 16 + l) * FEAT + 2 * half;
    const float* Brow = ppad + (size_t)(tj * 16 + l) * FEAT + 2 * half;

    v8f acc = {};
    for (int k = 0; k < FEAT; k += 4) {
        v2f a = *(const v2f*)(Arow + k);
        v2f b = *(const v2f*)(Brow + k);
        acc = __builtin_amdgcn_wmma_f32_16x16x4_f32(
            /*neg_a=*/false, a, /*neg_b=*/false, b,
            /*c_mod=*/(short)0, acc, /*reuse_a=*/false, /*reuse_b=*/false);
    }

    // C/D layout: lane<16 -> (M=r, N=lane); lane>=16 -> (M=r+8, N=lane-16)
    float local = 0.f;
    #pragma unroll
    for (int r = 0; r < 8; ++r) {
        int m  = r + half * 8;
        int ga = ti * 16 + m;        // global class index a
        int gb = tj * 16 + l;        // global class index b
        float g  = acc[r];
        float w  = (ga == gb) ? 0.f : cnts[ga] * cnts[gb];  // padded rows have cnt=0
        float d2 = sq[ga] + sq[gb] - 2.f * g;
        float dd = sqrtf(fmaxf(d2, 1e-12f));
        float h  = fmaxf(MARGIN_F - dd, 0.f);
        local += w * h * h;
    }
    for (int off = 16; off > 0; off >>= 1)
        local += __shfl_xor(local, off, 32);
    if (lane == 0) pl3[tile] = local;
}

// ---- fixed-order final reduction -> scalar loss ----
__global__ void final_kernel(const float* __restrict__ pl1, const float* __restrict__ pl2,
                             const float* __restrict__ pl3, float* __restrict__ out) {
    __shared__ float sm[256];
    int t = threadIdx.x;
    float s12 = 0.f;
    for (int i = t; i < CPAD; i += 256) s12 += pl1[i] + pl2[i];
    sm[t] = s12; __syncthreads();
    for (int o = 128; o > 0; o >>= 1) { if (t < o) sm[t] += sm[t + o]; __syncthreads(); }
    float tot12 = sm[0]; __syncthreads();
    float s3 = 0.f;
    for (int i = t; i < NTILES; i += 256) s3 += pl3[i];
    sm[t] = s3; __syncthreads();
    for (int o = 128; o > 0; o >>= 1) { if (t < o) sm[t] += sm[t + o]; __syncthreads(); }
    if (t == 0) {
        float inv12 = 1.f / ((float)NSAMP * (float)FEAT);   // mean over N*D
        float inv3  = 1.f / ((float)NSAMP * (float)NSAMP);  // mean over N*N
        out[0] = tot12 * inv12 + sm[0] * inv3;
    }
}

extern "C" void kernel_launch(void* const* d_in, const int* in_sizes, int n_in,
                              void* d_out, int out_size, void* d_ws, size_t ws_size,
                              hipStream_t stream) {
    (void)in_sizes; (void)n_in; (void)out_size; (void)ws_size;
    const float* m1 = (const float*)d_in[0];
    const float* m2 = (const float*)d_in[1];
    const float* cp = (const float*)d_in[2];
    const int*   tg = (const int*)d_in[3];
    float* out = (float*)d_out;
    float* ws  = (float*)d_ws;

    float* ppad = ws + OFF_PPAD;
    float* sq   = ws + OFF_SQ;
    float* cnts = ws + OFF_CNT;
    float* pl1  = ws + OFF_PL1;
    float* pl2  = ws + OFF_PL2;
    float* pl3  = ws + OFF_PL3;

    pad_kernel  <<<(CPAD * FEAT) / 256, 256, 0, stream>>>(cp, ppad);
    sq_kernel   <<<CPAD, 256, 0, stream>>>(ppad, sq);
    class_kernel<<<CPAD, 256, 0, stream>>>(m1, m2, cp, tg, cnts, pl1, pl2);
    gram_kernel <<<NTILES, 32, 0, stream>>>(ppad, sq, cnts, pl3);
    final_kernel<<<1, 256, 0, stream>>>(pl1, pl2, pl3, out);
}